// MyLayer_16801912062205
// MI455X (gfx1250) — compile-verified
//
#include <hip/hip_runtime.h>
#include <math.h>

// Problem constants (fixed by the reference)
#define KCLS   8
#define BATCH  256
#define NPTS   4096          // H*W
#define TILE   256           // points per block
#define NTILES (NPTS / TILE) // 16

typedef float v2f __attribute__((ext_vector_type(2)));
typedef float v8f __attribute__((ext_vector_type(8)));
typedef int   v4i __attribute__((ext_vector_type(4)));

typedef __attribute__((address_space(1))) v4i gv4i_t;  // global <4 x i32>
typedef __attribute__((address_space(3))) v4i lv4i_t;  // LDS    <4 x i32>

// (2*pi)^(-1.5)
#define INV_2PI_POW15 0.06349363593424097f

// ---------------- CDNA5 async global->LDS copy (16B per lane) ----------------
__device__ __forceinline__ void async_b128(const void* g, const void* l) {
#if __has_builtin(__builtin_amdgcn_global_load_async_to_lds_b128)
  __builtin_amdgcn_global_load_async_to_lds_b128(
      (gv4i_t*)(unsigned long long)(uintptr_t)g,
      (lv4i_t*)(unsigned)(uintptr_t)l, 0, 0);
#else
  asm volatile("global_load_async_to_lds_b128 %0, %1, off"
               :: "v"((unsigned)(uintptr_t)l), "v"(g)
               : "memory");
#endif
}

__device__ __forceinline__ void wait_async0() {
#if __has_builtin(__builtin_amdgcn_s_wait_asynccnt)
  __builtin_amdgcn_s_wait_asynccnt(0);
#else
  asm volatile("s_wait_asynccnt 0" ::: "memory");
#endif
}

// Fast 1-ulp reciprocal (v_rcp_f32) for uniform per-class constants.
__device__ __forceinline__ float frcp(float v) {
#if __has_builtin(__builtin_amdgcn_rcpf)
  return __builtin_amdgcn_rcpf(v);
#else
  return 1.0f / v;
#endif
}

// -----------------------------------------------------------------------------
__global__ __launch_bounds__(256) void gmm_density_kernel(
    const float* __restrict__ x,      // (B, N, 3)
    const float* __restrict__ scale,  // (K, 3, 3)
    const float* __restrict__ mean,   // (K, N, 3)
    float* __restrict__ out)          // flat (K, C, B, N)
{
  // sm[0:768)              : x tile   (256 pts * 3)
  // sm[768 + k*768 ...)    : mean tile for class k
  __shared__ float sm[768 + KCLS * 768];

  const int tid = threadIdx.x;
  const int b   = blockIdx.x >> 4;   // 0..255
  const int nt  = blockIdx.x & 15;   // tile within N
  const int n0  = nt * TILE;
  const int n   = n0 + tid;

  // ---- async stage: x tile (192 x b128) + 8 mean tiles (8*192 x b128) ----
  const float* xg = x + ((size_t)b * NPTS + n0) * 3;   // 768 contiguous floats
  if (tid < 192) {
    async_b128(xg + tid * 4, &sm[tid * 4]);
  }
  for (int i = tid; i < KCLS * 192; i += 256) {
    const int k = i / 192;
    const int c = i - k * 192;
    const float* mg = mean + ((size_t)k * NPTS + n0) * 3 + c * 4;
    async_b128(mg, &sm[768 + k * 768 + c * 4]);
  }
  wait_async0();
  __syncthreads();

  const float x0 = sm[tid * 3 + 0];
  const float x1 = sm[tid * 3 + 1];
  const float x2 = sm[tid * 3 + 2];

  const unsigned lane = tid & 31u;
  const unsigned row  = lane & 15u;   // M-row (A) / N-column (B,D) index
  const bool     lo   = lane < 16u;

#pragma unroll 2
  for (int k = 0; k < KCLS; ++k) {
    // ---- lower-triangular Cholesky factor (uniform -> scalar loads) ----
    const float L00 = scale[k * 9 + 0];
    const float L10 = scale[k * 9 + 3];
    const float L11 = scale[k * 9 + 4];
    const float L20 = scale[k * 9 + 6];
    const float L21 = scale[k * 9 + 7];
    const float L22 = scale[k * 9 + 8];
    const float i00 = frcp(L00);
    const float i11 = frcp(L11);
    const float i22 = frcp(L22);
    const float i10 = -L10 * i00 * i11;
    const float i21 = -L21 * i11 * i22;
    const float i20 = (L10 * L21 - L20 * L11) * (i00 * i11 * i22);
    // |det L^{-1}| = |i00*i11*i22|  ->  no extra divide for the normalizer
    const float norm = INV_2PI_POW15 * fabsf(i00 * i11 * i22);

    // ---- per-point diff from LDS ----
    const float* mu = &sm[768 + k * 768 + tid * 3];
    const float d0 = x0 - mu[0];
    const float d1 = x1 - mu[1];
    const float d2 = x2 - mu[2];

    float maha;
#if __has_builtin(__builtin_amdgcn_wmma_f32_16x16x4_f32)
    // A (16x4 f32): lanes 0-15 hold row M=lane, VGPR0=K0, VGPR1=K1;
    //               lanes 16-31 hold row M=lane-16, VGPR0=K2, VGPR1=K3(=0).
    float a0, a1;
    if (lo) {
      a0 = (row == 0u) ? i00 : (row == 1u) ? i10 : (row == 2u) ? i20 : 0.0f;
      a1 = (row == 1u) ? i11 : (row == 2u) ? i21 : 0.0f;
    } else {
      a0 = (row == 2u) ? i22 : 0.0f;
      a1 = 0.0f;
    }
    const v2f A = {a0, a1};

    // B (4x16 f32): lanes 0-15: VGPR0=rowK0(d0), VGPR1=rowK1(d1), N=lane;
    //               lanes 16-31: VGPR0=rowK2(d2), VGPR1=rowK3(=0), N=lane-16.
    // WMMA #1 covers the wave's points 0..15, #2 covers points 16..31.
    const float s2  = __shfl(d2, (int)(lane & 15u));        // d2 of point lane-16
    const v2f B1 = {lo ? d0 : s2, lo ? d1 : 0.0f};
    v8f acc1 = {};
    acc1 = __builtin_amdgcn_wmma_f32_16x16x4_f32(
        false, A, false, B1, (short)0, acc1, false, false);
    const float m1 = acc1[0] * acc1[0] + acc1[1] * acc1[1] + acc1[2] * acc1[2];

    const float t0 = __shfl(d0, (int)(16u + (lane & 15u)));
    const float t1 = __shfl(d1, (int)(16u + (lane & 15u)));
    const float t2 = __shfl(d2, (int)(16u + (lane & 15u)));
    const v2f B2 = {lo ? t0 : t2, lo ? t1 : 0.0f};
    v8f acc2 = {};
    acc2 = __builtin_amdgcn_wmma_f32_16x16x4_f32(
        false, A, false, B2, (short)0, acc2, false, false);
    const float m2  = acc2[0] * acc2[0] + acc2[1] * acc2[1] + acc2[2] * acc2[2];
    const float m2s = __shfl(m2, (int)(lane & 15u));        // move to hi lanes
    maha = lo ? m1 : m2s;
#else
    const float y0 = i00 * d0;
    const float y1 = i10 * d0 + i11 * d1;
    const float y2 = i20 * d0 + i21 * d1 + i22 * d2;
    maha = y0 * y0 + y1 * y1 + y2 * y2;
#endif

    const float g = norm * __expf(-0.5f * maha);

    // out[((k*3+c)*B + b)*N + n]; contiguous in n across lanes -> coalesced.
    // Write-once streaming output: non-temporal to preserve L2 for mean/x.
    __builtin_nontemporal_store(
        g * x0, out + ((size_t)(k * 3 + 0) * BATCH + b) * NPTS + n);
    __builtin_nontemporal_store(
        g * x1, out + ((size_t)(k * 3 + 1) * BATCH + b) * NPTS + n);
    __builtin_nontemporal_store(
        g * x2, out + ((size_t)(k * 3 + 2) * BATCH + b) * NPTS + n);
  }
}

extern "C" void kernel_launch(void* const* d_in, const int* in_sizes, int n_in,
                              void* d_out, int out_size, void* d_ws, size_t ws_size,
                              hipStream_t stream) {
  (void)in_sizes; (void)n_in; (void)out_size; (void)d_ws; (void)ws_size;
  const float* x     = (const float*)d_in[0];
  const float* scale = (const float*)d_in[1];
  const float* mean  = (const float*)d_in[2];
  float* out = (float*)d_out;

  dim3 grid(BATCH * NTILES);  // 4096 blocks: (b, n-tile)
  dim3 block(256);            // 8 wave32 per block
  gmm_density_kernel<<<grid, block, 0, stream>>>(x, scale, mean, out);
}